// RNNBase_48026324304051
// MI455X (gfx1250) — compile-verified
//
#include <hip/hip_runtime.h>
#include <hip/hip_bf16.h>

// ---------------------------------------------------------------------------
// 2-layer LSTM forward for MI455X (gfx1250), fp32 via V_WMMA_F32_16X16X4_F32.
//   T=2048, B=32, I=H=512, G=2048, gate order i,f,g,o.
//   * WMMA-ready packed layouts: every GEMM operand is one aligned b64 load,
//     P is stored tile-major in the C/D lane layout (b128 traffic only).
//   * per 512-step chunk: proj_k (throughput GEMM, M=2 register blocking)
//     then rnn_step_k (32 persistent blocks, 1 grid barrier/step).
// ---------------------------------------------------------------------------

#define T_     2048
#define B_     32
#define H_     512
#define G_     2048
#define K4_    (H_ / 4)     // 128 packed K groups
#define CHUNK  512
#define NCHUNK (T_ / CHUNK)
#define BH     (B_ * H_)
#define YSZ    ((size_t)T_ * B_ * H_)

typedef float v2f __attribute__((ext_vector_type(2)));
typedef float v4f __attribute__((ext_vector_type(4)));
typedef float v8f __attribute__((ext_vector_type(8)));

__device__ __forceinline__ v8f wmma_f32(v2f a, v2f b, v8f c) {
  // D = A(16x4) * B(4x16) + C(16x16), native fp32 matrix op
  return __builtin_amdgcn_wmma_f32_16x16x4_f32(
      false, a, false, b, (short)0, c, false, false);
}

__device__ __forceinline__ float sigmoidf_(float x) {
  return 1.0f / (1.0f + __expf(-x));
}

// ------------------- pack W[G][512] -> Wp[k4][G][4] ------------------------
// One float4 (4 consecutive K) per (k4,g); coalesced b128 writes.
__global__ __launch_bounds__(256) void pack_w_k(const float* __restrict__ W,
                                                float* __restrict__ Wp) {
  int i = blockIdx.x * 256 + threadIdx.x;  // i = k4*G + g
  if (i < K4_ * G_) {
    int k4 = i / G_;
    int g  = i % G_;
    v4f v = *(const v4f*)&W[(size_t)g * H_ + k4 * 4];
    *(v4f*)&Wp[(size_t)i * 4] = v;
  }
}

// ---------------- initial state: hP[k4][B][4], c[b][h] ---------------------
__global__ __launch_bounds__(256) void init_state_k(const float* __restrict__ h0,
                                                    const float* __restrict__ c0,
                                                    float* __restrict__ hP,
                                                    float* __restrict__ cst) {
  int i = blockIdx.x * 256 + threadIdx.x;  // over B*H
  if (i < BH) {
    int b = i / H_, h = i % H_;
    hP[(h >> 2) * (B_ * 4) + b * 4 + (h & 3)] = h0[i];
    cst[i] = c0[i];
  }
}

// ---------------- proj: Pp = X . W^T + (b_ih + b_hh), tile-major -----------
// Block covers 32 rows (2 M-tiles), packed in 64KB LDS. 8 waves x 16 N-tiles;
// each wave reuses one B b64 across 2 WMMAs (M register blocking).
__global__ __launch_bounds__(256) void proj_k(const float* __restrict__ X,   // [rows][512]
                                              const float* __restrict__ Wp,  // [k4][G][4]
                                              const float* __restrict__ b1,
                                              const float* __restrict__ b2,
                                              float* __restrict__ Pp) {      // [mt][nt][lane][8]
  __shared__ float Ap[K4_ * 32 * 4];  // 64 KB: [k4][m(32)][koff]
  const int tid = threadIdx.x;
  const int m0  = blockIdx.x * 32;

  for (int i = tid; i < 32 * 512; i += 256) {
    int m = i >> 9;
    int k = i & 511;
    Ap[(k >> 2) * 128 + m * 4 + (k & 3)] = X[(size_t)(m0 + m) * 512 + k];
  }
  __syncthreads();

  const int wave = tid >> 5;
  const int lane = tid & 31;
  const int l15  = lane & 15;
  const int hi   = lane >> 4;                 // 0: K+0/1, 1: K+2/3
  const int a_off0 = l15 * 4 + hi * 2;        // m-tile 0
  const int a_off1 = (16 + l15) * 4 + hi * 2; // m-tile 1
  const int mt_base = blockIdx.x * 2;

  for (int nti = 0; nti < 16; ++nti) {
    const int nt = wave * 16 + nti;
    const int g0 = nt * 16;
    const size_t b_base = (size_t)(g0 + l15) * 4 + hi * 2;
    const float bias = b1[g0 + l15] + b2[g0 + l15];
    v8f c0, c1;
#pragma unroll
    for (int j = 0; j < 8; ++j) { c0[j] = bias; c1[j] = bias; }
#pragma unroll 8
    for (int k4 = 0; k4 < K4_; ++k4) {
      v2f b  = *(const v2f*)&Wp[(size_t)k4 * (G_ * 4) + b_base];
      v2f a0 = *(const v2f*)&Ap[k4 * 128 + a_off0];
      v2f a1 = *(const v2f*)&Ap[k4 * 128 + a_off1];
      c0 = wmma_f32(a0, b, c0);
      c1 = wmma_f32(a1, b, c1);
    }
    *(v8f*)(Pp + (((size_t)(mt_base + 0) * 128) + nt) * 256 + lane * 8) = c0;
    *(v8f*)(Pp + (((size_t)(mt_base + 1) * 128) + nt) * 256 + lane * 8) = c1;
  }
}

// ----------------------------- grid barrier --------------------------------
__device__ __forceinline__ void gsync(unsigned* cnt, unsigned target) {
  __threadfence();
  __syncthreads();
  if (threadIdx.x == 0) {
    atomicAdd(cnt, 1u);
    while (__hip_atomic_load(cnt, __ATOMIC_ACQUIRE, __HIP_MEMORY_SCOPE_AGENT) < target)
      __builtin_amdgcn_s_sleep(2);
  }
  __syncthreads();
  __threadfence();
}

// ------------------------------ recurrence ---------------------------------
// 32 persistent blocks; block owns 16 h-columns. 8 waves = {i,f,g,o} x {2 b-tiles},
// each computes one 16x16 gate tile: C = P_t + h_t . W_hh^T (C seeded from Pp).
__global__ __launch_bounds__(256) void rnn_step_k(
    const float* __restrict__ Pp,   // tile-major [t*2+r][q*32+bid][lane][8]
    const float* __restrict__ Wp,   // [k4][G][4] packed W_hh^T
    float* __restrict__ y,          // [CHUNK][B][H] (caller offsets)
    float* __restrict__ hP0,        // ping  [k4][B][4]
    float* __restrict__ hP1,        // pong
    float* __restrict__ cst,        // [B][H]
    unsigned* __restrict__ sync_cnt,
    int write_final,
    float* __restrict__ hn,         // [B][H]
    float* __restrict__ cn) {       // [B][H]
  __shared__ float lds_g[8 * 256];  // 8 gate tiles of 16x16

  const int tid  = threadIdx.x;
  const int bid  = blockIdx.x;      // h-columns [bid*16, bid*16+16)
  const int wave = tid >> 5;
  const int lane = tid & 31;
  const int q    = wave >> 1;       // gate: 0=i 1=f 2=g 3=o
  const int r    = wave & 1;        // b-tile
  const int l15  = lane & 15;
  const int hi   = lane >> 4;
  const int a_off  = (r * 16 + l15) * 4 + hi * 2;        // hP[k4*128 + .]
  const size_t b_base = (size_t)(q * H_ + bid * 16 + l15) * 4 + hi * 2;
  const int ntile = q * 32 + bid;
  float* hbufs[2] = {hP0, hP1};

  for (int t = 0; t < CHUNK; ++t) {
    const float* hT  = hbufs[t & 1];
    float*       hTn = hbufs[(t + 1) & 1];

    // seed accumulator with precomputed input projection (+biases): 2x b128
    const float* Ps = Pp + (((size_t)t * 2 + r) * 128 + ntile) * 256 + lane * 8;
    v8f c = *(const v8f*)Ps;

    if (t + 1 < CHUNK)  // warm next step's P tile into cache
      __builtin_prefetch(Ps + 2 * 128 * 256, 0, 1);

    // K reduction over H=512, one b64 per operand per WMMA (128 fp32 WMMAs)
#pragma unroll 8
    for (int k4 = 0; k4 < K4_; ++k4) {
      v2f a = *(const v2f*)&hT[k4 * (B_ * 4) + a_off];
      v2f b = *(const v2f*)&Wp[(size_t)k4 * (G_ * 4) + b_base];
      c = wmma_f32(a, b, c);
    }

    // publish gate tile to LDS
#pragma unroll
    for (int j = 0; j < 8; ++j)
      lds_g[wave * 256 + (hi * 8 + j) * 16 + l15] = c[j];
    __syncthreads();

    // elementwise LSTM cell update: 512 elements (32 b x 16 cols)
    for (int e = tid; e < B_ * 16; e += 256) {
      const int col = e & 15;
      const int b   = e >> 4;
      const int rr  = b >> 4;
      const int m   = b & 15;
      float iv = lds_g[(0 * 2 + rr) * 256 + m * 16 + col];
      float fv = lds_g[(1 * 2 + rr) * 256 + m * 16 + col];
      float gv = lds_g[(2 * 2 + rr) * 256 + m * 16 + col];
      float ov = lds_g[(3 * 2 + rr) * 256 + m * 16 + col];
      iv = sigmoidf_(iv);
      fv = sigmoidf_(fv);
      gv = tanhf(gv);
      ov = sigmoidf_(ov);
      const int hcol = bid * 16 + col;
      const float cv = fv * cst[b * H_ + hcol] + iv * gv;
      cst[b * H_ + hcol] = cv;
      const float hv = ov * tanhf(cv);
      hTn[(hcol >> 2) * (B_ * 4) + b * 4 + (hcol & 3)] = hv;
      y[((size_t)t * B_ + b) * H_ + hcol] = hv;
      if (write_final && t == CHUNK - 1) {
        hn[b * H_ + hcol] = hv;
        cn[b * H_ + hcol] = cv;
      }
    }

    // device-wide barrier (one per step; h is ping-ponged)
    gsync(sync_cnt, (unsigned)((t + 1) * gridDim.x));
  }
}

// ---------------------------------------------------------------------------
extern "C" void kernel_launch(void* const* d_in, const int* in_sizes, int n_in,
                              void* d_out, int out_size, void* d_ws, size_t ws_size,
                              hipStream_t stream) {
  (void)in_sizes; (void)n_in; (void)out_size; (void)ws_size;

  const float* x    = (const float*)d_in[0];
  const float* h0   = (const float*)d_in[1];   // [2,B,H]
  const float* c0   = (const float*)d_in[2];
  const float* Wih0 = (const float*)d_in[3];
  const float* Whh0 = (const float*)d_in[4];
  const float* bih0 = (const float*)d_in[5];
  const float* bhh0 = (const float*)d_in[6];
  const float* Wih1 = (const float*)d_in[7];
  const float* Whh1 = (const float*)d_in[8];
  const float* bih1 = (const float*)d_in[9];
  const float* bhh1 = (const float*)d_in[10];
  float* out = (float*)d_out;

  // workspace layout (floats)
  float* Pp   = (float*)d_ws;          // CHUNK*B*G = 33,554,432
  float* y0   = Pp   + (size_t)CHUNK * B_ * G_;
  float* WpI0 = y0   + YSZ;            // 4 x packed [128][2048][4] weights
  float* WpH0 = WpI0 + (size_t)H_ * G_;
  float* WpI1 = WpH0 + (size_t)H_ * G_;
  float* WpH1 = WpI1 + (size_t)H_ * G_;
  float* hPa  = WpH1 + (size_t)H_ * G_;
  float* hPb  = hPa + BH;
  float* cst  = hPb + BH;
  unsigned* sync = (unsigned*)(cst + BH);

  const int npack = (K4_ * G_ + 255) / 256;
  pack_w_k<<<npack, 256, 0, stream>>>(Wih0, WpI0);
  pack_w_k<<<npack, 256, 0, stream>>>(Whh0, WpH0);
  pack_w_k<<<npack, 256, 0, stream>>>(Wih1, WpI1);
  pack_w_k<<<npack, 256, 0, stream>>>(Whh1, WpH1);

  const float* xin[2]  = {x, y0};
  float*       yout[2] = {y0, out};
  const float* WpI[2]  = {WpI0, WpI1};
  const float* WpH[2]  = {WpH0, WpH1};
  const float* bi[2]   = {bih0, bih1};
  const float* bh[2]   = {bhh0, bhh1};

  for (int L = 0; L < 2; ++L) {
    init_state_k<<<(BH + 255) / 256, 256, 0, stream>>>(h0 + (size_t)L * BH,
                                                       c0 + (size_t)L * BH, hPa, cst);
    for (int ch = 0; ch < NCHUNK; ++ch) {
      const size_t row0 = (size_t)ch * CHUNK * B_;
      proj_k<<<CHUNK * B_ / 32, 256, 0, stream>>>(xin[L] + row0 * H_, WpI[L],
                                                  bi[L], bh[L], Pp);
      hipMemsetAsync(sync, 0, 64, stream);
      rnn_step_k<<<32, 256, 0, stream>>>(
          Pp, WpH[L], yout[L] + row0 * H_, hPa, hPb, cst, sync,
          /*write_final=*/(ch == NCHUNK - 1) ? 1 : 0,
          out + YSZ + (size_t)L * BH,
          out + YSZ + 2 * (size_t)BH + (size_t)L * BH);
    }
  }
}